// NetAttention_18829136626259
// MI455X (gfx1250) — compile-verified
//
#include <hip/hip_runtime.h>
#include <stdint.h>

typedef __attribute__((ext_vector_type(2))) float v2f;
typedef __attribute__((ext_vector_type(8))) float v8f;

#define BSZ 2048
#define KN  32
#define DD  256
#define VSTRIDE 260   // pad so per-lane row gathers hit distinct LDS banks
#define ASTRIDE 34    // even -> 8B-aligned float2 rows

__device__ __forceinline__ void wait_asynccnt0() {
#if __has_builtin(__builtin_amdgcn_s_wait_asynccnt)
  __builtin_amdgcn_s_wait_asynccnt(0);
#else
  asm volatile("s_wait_asynccnt 0x0" ::: "memory");
#endif
}

// ---------------------------------------------------------------------------
// Kernel 1: proj = node_flat(2048x256) @ W(256x256), fp32 WMMA 16x16x4.
// One 16x16 C tile per wave; 4 waves / block; 2048 tiles -> 512 blocks.
// ---------------------------------------------------------------------------
__global__ __launch_bounds__(128) void proj_gemm_kernel(
    const float* __restrict__ node, const float* __restrict__ W,
    float* __restrict__ proj) {
  const int lane = threadIdx.x & 31;
  const int wave = threadIdx.x >> 5;
  const int tile = blockIdx.x * 4 + wave;       // 0..2047
  const int tm = tile >> 4;                     // 0..127 (row tile)
  const int tn = tile & 15;                     // 0..15  (col tile)
  const int row0 = tm * 16, col0 = tn * 16;
  const int half = lane >> 4;                   // 0: K lo pair, 1: K hi pair
  const int lid  = lane & 15;

  const float* arow = node + (size_t)(row0 + lid) * DD;
  v8f c = {};
  for (int k0 = 0; k0 < DD; k0 += 4) {
    const int ka = k0 + 2 * half;
    v2f a = *(const v2f*)(arow + ka);           // A[m][ka], A[m][ka+1]
    v2f bb;
    bb.x = W[(size_t)ka * DD + col0 + lid];     // B[ka][n]
    bb.y = W[(size_t)(ka + 1) * DD + col0 + lid];
    c = __builtin_amdgcn_wmma_f32_16x16x4_f32(false, a, false, bb,
                                              (short)0, c, false, false);
  }
  const int col = col0 + lid;
#pragma unroll
  for (int r = 0; r < 8; ++r) {
    proj[(size_t)(row0 + r + 8 * half) * DD + col] = c[r];
  }
}

// ---------------------------------------------------------------------------
// Kernel 2: per batch row b (one block each):
//   Q = proj[32*(b%64) .. +32]  (read from global; L2-hot)
//   V = neigh[b]                (staged once into LDS via async global->LDS)
//   S = Q V^T + bias[n];  A = softmax_rows(S);  O = A V
// ---------------------------------------------------------------------------
__global__ __launch_bounds__(128) void attn_kernel(
    const float* __restrict__ neigh, const float* __restrict__ proj,
    const float* __restrict__ bias, float* __restrict__ outO,
    float* __restrict__ outA) {
  __shared__ float Vs[KN][VSTRIDE];
  __shared__ float Ss[KN][KN + 1];
  __shared__ float As[KN][ASTRIDE];

  const int b    = blockIdx.x;
  const int tid  = threadIdx.x;
  const int lane = tid & 31;
  const int wave = tid >> 5;
  const int half = lane >> 4;
  const int lid  = lane & 15;

  const float* vsrc = neigh + (size_t)b * KN * DD;
  const float* qsrc = proj + (size_t)(32 * (b & 63)) * DD;

  // Stage V (32x256 fp32 = 32KB) straight into LDS with async b128 copies
  // (ASYNCcnt path; bypasses the VGPR round-trip). 2048 16B chunks / 128 thr.
  for (int i = tid; i < KN * (DD / 4); i += 128) {
    const int row = i >> 6;                     // 64 float4 per row
    const int c4  = (i & 63) * 4;
    const uint64_t gaddr = (uint64_t)(uintptr_t)(vsrc + (size_t)row * DD + c4);
    // Flat-aperture mapping: LDS byte address = low 32 bits of generic ptr.
    const uint32_t laddr = (uint32_t)(uintptr_t)(&Vs[row][c4]);
    asm volatile("global_load_async_to_lds_b128 %0, %1, off"
                 :
                 : "v"(laddr), "v"(gaddr)
                 : "memory");
  }
  wait_asynccnt0();
  __syncthreads();

  // ---- S = Q V^T : wave -> 16x16 tile (mt = wave>>1, nt = wave&1) ----
  {
    const int mt = wave >> 1, nt = wave & 1;
    const float* qrow = qsrc + (size_t)(mt * 16 + lid) * DD;
    v8f c = {};
    for (int d0 = 0; d0 < DD; d0 += 4) {
      const int ka = d0 + 2 * half;
      v2f a  = *(const v2f*)(qrow + ka);              // Q[m][ka..ka+1]
      v2f bb = *(const v2f*)(&Vs[nt * 16 + lid][ka]); // (V^T)[ka..][n] = V[n][ka..]
      c = __builtin_amdgcn_wmma_f32_16x16x4_f32(false, a, false, bb,
                                                (short)0, c, false, false);
    }
    const int col = nt * 16 + lid;
    const float bv = bias[col];                  // bias[n] broadcasts over rows m
#pragma unroll
    for (int r = 0; r < 8; ++r)
      Ss[mt * 16 + r + 8 * half][col] = c[r] + bv;
  }
  __syncthreads();

  // ---- row softmax (32 rows, one thread per row) + write A out ----
  if (tid < KN) {
    float m = -3.402823466e38f;
#pragma unroll
    for (int n = 0; n < KN; ++n) m = fmaxf(m, Ss[tid][n]);
    float e[KN];
    float s = 0.f;
#pragma unroll
    for (int n = 0; n < KN; ++n) { e[n] = __expf(Ss[tid][n] - m); s += e[n]; }
    const float inv = 1.f / s;
    float* arow = outA + (size_t)b * KN * KN + (size_t)tid * KN;
#pragma unroll
    for (int n = 0; n < KN; ++n) {
      const float av = e[n] * inv;
      As[tid][n] = av;
      arow[n] = av;
    }
  }
  __syncthreads();

  // ---- O = A V : 32 tiles (2 x 16); each wave does 8 tiles ----
  float* obase = outO + (size_t)b * KN * DD;
  for (int t = wave; t < 32; t += 4) {
    const int mt = t >> 4, nt = t & 15;
    v8f c = {};
#pragma unroll
    for (int k0 = 0; k0 < KN; k0 += 4) {
      const int ka = k0 + 2 * half;
      v2f a = *(const v2f*)(&As[mt * 16 + lid][ka]);  // A[m][ka..ka+1]
      v2f bb;
      bb.x = Vs[ka][nt * 16 + lid];                   // V[ka][n]
      bb.y = Vs[ka + 1][nt * 16 + lid];
      c = __builtin_amdgcn_wmma_f32_16x16x4_f32(false, a, false, bb,
                                                (short)0, c, false, false);
    }
    const int col = nt * 16 + lid;
#pragma unroll
    for (int r = 0; r < 8; ++r)
      obase[(size_t)(mt * 16 + r + 8 * half) * DD + col] = c[r];
  }
}

// ---------------------------------------------------------------------------
extern "C" void kernel_launch(void* const* d_in, const int* in_sizes, int n_in,
                              void* d_out, int out_size, void* d_ws, size_t ws_size,
                              hipStream_t stream) {
  const float* node  = (const float*)d_in[0];  // (B,T,H) = (2048,256) flat
  const float* neigh = (const float*)d_in[1];  // (B,K,T,H) = (2048,32,256) flat
  const float* W     = (const float*)d_in[2];  // (256,256)
  const float* bias  = (const float*)d_in[3];  // (32,)
  (void)in_sizes; (void)n_in; (void)out_size; (void)ws_size;

  float* proj = (float*)d_ws;                  // 2048*256 fp32 = 2 MB scratch
  float* outO = (float*)d_out;                 // 2048*32*256 floats
  float* outA = outO + (size_t)BSZ * KN * DD;  // then 2048*32*32 floats

  proj_gemm_kernel<<<512, 128, 0, stream>>>(node, W, proj);
  attn_kernel<<<BSZ, KN * 4, 0, stream>>>(neigh, proj, bias, outO, outA);
}